// ImprovedAttention_53034256171069
// MI455X (gfx1250) — compile-verified
//
#include <hip/hip_runtime.h>
#include <hip/hip_bf16.h>
#include <math.h>

// ---------------------------------------------------------------------------
// Quad-split 2D Mamba block for MI455X (gfx1250, wave32, WMMA + TDM).
//   x: (2, 96, 128, 128) f32 -> 4 quadrants of 64x64, each an independent
//   Mamba block.  Per quadrant: B=2, L=4096, M=8192 rows.
// ---------------------------------------------------------------------------

#define D_MODEL 96
#define D_INNER 192
#define D_STATE 16
#define DT_RANK 6
#define QH 64
#define QW 64
#define LQ (QH * QW)          // 4096
#define BATCH 2
#define MROWS (BATCH * LQ)    // 8192 rows per quadrant

// workspace layout (floats), per quadrant:
//   [0        , QF)      x_in   (aliased later as y)
//   [QF       , 2QF)     res
//   [2QF      , 3QF)     xc
//   [3QF      , 3QF+XD)  x_dbl (padded 38 -> 48)
#define QF   ((size_t)MROWS * D_INNER)   // 1,572,864
#define XDP  48
#define XDSZ ((size_t)MROWS * XDP)       // 393,216
#define PERQ (3 * QF + XDSZ)

typedef float v2f __attribute__((ext_vector_type(2)));
typedef float v8f __attribute__((ext_vector_type(8)));
typedef unsigned int v4u __attribute__((ext_vector_type(4)));
typedef int v4i __attribute__((ext_vector_type(4)));
typedef int v8i __attribute__((ext_vector_type(8)));

struct QParams {
    const float* in_proj;   // (96,384)
    const float* conv_w;    // (192,4)
    const float* conv_b;    // (192,)
    const float* x_proj;    // (192,38)
    const float* dt_proj;   // (6,192)
    const float* dt_bias;   // (192,)
    const float* A_log;     // (192,16)
    const float* D;         // (192,)
    const float* out_proj;  // (192,96)
};
struct AllParams { QParams q[4]; };

__device__ __forceinline__ float silu_f(float v) {
    return v / (1.0f + __expf(-v));
}

// ---------------------------------------------------------------------------
// Tensor Data Mover descriptor helpers (CDNA5 ISA ch.8, D# groups).
// group0: [1:0]=count=1, [63:32]=lds_addr, [120:64]=global_addr, [127:126]=2
// group1: [17:16]=data_size(2 -> 4B), [79:48]=tensor_dim0, [111:80]=tensor_dim1,
//         [127:112]=tile_dim0, [143:128]=tile_dim1, [207:160]=dim0_stride
// ---------------------------------------------------------------------------
#if __has_builtin(__builtin_amdgcn_tensor_load_to_lds)
#define HAVE_TDM 1

__device__ __forceinline__ v4u tdm_g0(unsigned lds_byte, unsigned long long ga) {
    v4u g;
    g.x = 1u;                                            // count=1, user mode
    g.y = lds_byte;                                      // lds_addr
    g.z = (unsigned)(ga & 0xffffffffu);                  // global_addr[31:0]
    g.w = (unsigned)((ga >> 32) & 0x01ffffffu) | 0x80000000u;  // [56:32] | type=2
    return g;
}

__device__ __forceinline__ v8i tdm_g1(unsigned dim0, unsigned dim1,
                                      unsigned tile0, unsigned tile1,
                                      unsigned stride0) {
    v8i g;
    g[0] = 0x20000;                                      // data_size = 2 (4 bytes)
    g[1] = (int)((dim0 & 0xffffu) << 16);                // tensor_dim0[15:0]
    g[2] = (int)(((dim0 >> 16) & 0xffffu) | ((dim1 & 0xffffu) << 16));
    g[3] = (int)(((dim1 >> 16) & 0xffffu) | ((tile0 & 0xffffu) << 16));
    g[4] = (int)(tile1 & 0xffffu);                       // tile_dim1 (tile_dim2=0)
    g[5] = (int)stride0;                                 // dim0_stride[31:0]
    g[6] = 0;
    g[7] = 0;
    return g;
}

__device__ __forceinline__ void tdm_load(v4u g0, v8i g1) {
    v4i z4 = {0, 0, 0, 0};
#if defined(__clang_major__) && __clang_major__ >= 23
    v8i z8 = {0, 0, 0, 0, 0, 0, 0, 0};
    __builtin_amdgcn_tensor_load_to_lds(g0, g1, z4, z4, z8, 0);
#else
    __builtin_amdgcn_tensor_load_to_lds(g0, g1, z4, z4, 0);
#endif
}
#endif

// ---------------------------------------------------------------------------
// K1: in_proj GEMM.  grid = (384/64, 8192/16, 4), block = 32 (one wave).
// Each wave computes a 16x64 tile: the gathered A-fragment is reused across
// 4 N-tiles (4x less A traffic).  f32 WMMA 16x16x4, K=96.
// ---------------------------------------------------------------------------
__global__ __launch_bounds__(32)
void k_in_proj(const float* __restrict__ x, AllParams P, float* __restrict__ ws) {
    const int q    = blockIdx.z;
    const int lane = threadIdx.x;
    const int m0   = blockIdx.y * 16;
    const int n0   = blockIdx.x * 64;

    const QParams& p = P.q[q];
    float* xin = ws + (size_t)q * PERQ;
    float* res = xin + QF;

    const int colb = n0 + (lane & 15);
    const int rowA = m0 + (lane & 15);
    const int kk   = (lane >> 4) * 2;        // lanes 0-15: K={0,1}; 16-31: K={2,3}

    const int oh = (q >> 1) * QH, ow = (q & 1) * QW;
    const int bA = rowA >> 12;
    const int lA = rowA & (LQ - 1);
    const int hA = oh + (lA >> 6);
    const int wA = ow + (lA & 63);
    const float* xrow = x + (((size_t)bA * D_MODEL) * 128 + hA) * 128 + wA;

    v8f acc[4];
    #pragma unroll
    for (int j = 0; j < 4; ++j) acc[j] = (v8f){0.f,0.f,0.f,0.f,0.f,0.f,0.f,0.f};

    #pragma unroll 4
    for (int k = 0; k < D_MODEL; k += 4) {
        v2f a;
        a.x = xrow[(size_t)(k + kk)     * 16384];
        a.y = xrow[(size_t)(k + kk + 1) * 16384];
        const float* w0 = p.in_proj + (k + kk) * 384 + colb;
        #pragma unroll
        for (int j = 0; j < 4; ++j) {
            v2f b;
            b.x = w0[16 * j];
            b.y = w0[384 + 16 * j];
            acc[j] = __builtin_amdgcn_wmma_f32_16x16x4_f32(
                         false, a, false, b, (short)0, acc[j], false, false);
        }
    }

    // n0 is a multiple of 64, so the 64-wide tile is entirely in x_in or res
    float* base = (n0 < D_INNER) ? xin : (res - D_INNER);
    const int mBase = m0 + ((lane >> 4) << 3);
    #pragma unroll
    for (int j = 0; j < 4; ++j) {
        float* dst = base + colb + 16 * j;
        #pragma unroll
        for (int r = 0; r < 8; ++r)
            dst[(size_t)(mBase + r) * D_INNER] = acc[j][r];
    }
}

// ---------------------------------------------------------------------------
// K2: depthwise causal conv(4) + SiLU -> xc (LDS + global), then
//     x_dbl = xc @ x_proj (N=38, padded to 48).  grid = (64, 2, 4), block 256.
// ---------------------------------------------------------------------------
#define CT 64            // rows of L per block
#define XCS 193          // LDS row stride (odd vs 64 banks)

__global__ __launch_bounds__(256)
void k_conv_xproj(AllParams P, float* __restrict__ ws) {
    __shared__ float xc_s[CT * XCS];

    const int q  = blockIdx.z;
    const int b  = blockIdx.y;
    const int l0 = blockIdx.x * CT;
    const int t  = threadIdx.x;

    const QParams& p = P.q[q];
    const float* xin  = ws + (size_t)q * PERQ;
    float*       xc   = ws + (size_t)q * PERQ + 2 * QF;
    float*       xdbl = ws + (size_t)q * PERQ + 3 * QF;
    const size_t rb   = (size_t)b * LQ;

    for (int idx = t; idx < CT * D_INNER; idx += 256) {
        const int r = idx / D_INNER;
        const int e = idx - r * D_INNER;
        const int l = l0 + r;
        float accv = p.conv_b[e];
        #pragma unroll
        for (int j = 0; j < 4; ++j) {
            const int ls = l - 3 + j;
            const float v = (ls >= 0) ? xin[(rb + ls) * D_INNER + e] : 0.f;
            accv = fmaf(p.conv_w[e * 4 + j], v, accv);
        }
        const float s = silu_f(accv);
        xc_s[r * XCS + e] = s;
        xc[(rb + l) * D_INNER + e] = s;
    }
    __syncthreads();

    for (int idx = t; idx < CT * 38; idx += 256) {
        const int r = idx / 38;
        const int n = idx - r * 38;
        const float* xr = xc_s + r * XCS;
        const float* wp = p.x_proj + n;      // stride 38
        float accv = 0.f;
        #pragma unroll 8
        for (int k = 0; k < D_INNER; ++k)
            accv = fmaf(xr[k], wp[k * 38], accv);
        xdbl[(rb + l0 + r) * XDP + n] = accv;
    }
}

// ---------------------------------------------------------------------------
// K3: selective scan.  grid = (192/8, 2, 4), block 128 (4 waves, 8 channels).
// lane%16 = state n; each half-wave owns one channel.  dt fused on the fly
// (dt_proj dot + softplus).  x_dbl/xc double-buffered in LDS via the Tensor
// Data Mover: wave 0 issues tensor_load_to_lds for chunk c+1 while all waves
// run the 32 serial steps of chunk c, then s_wait_tensorcnt 0 + barrier.
// ---------------------------------------------------------------------------
#define SCH 32           // steps per LDS chunk

__global__ __launch_bounds__(128)
void k_scan(AllParams P, float* __restrict__ ws) {
    __shared__ float xd_s[2][SCH * XDP];   // 2 x (32 rows x 48)
    __shared__ float xc_s[2][SCH * 8];
    __shared__ float y_s [SCH * 8];

    const int q  = blockIdx.z;
    const int b  = blockIdx.y;
    const int e0 = blockIdx.x * 8;
    const int t  = threadIdx.x;
    const int lane = t & 31;
    const int ch   = t >> 4;             // 0..7
    const int e    = e0 + ch;
    const int n    = lane & 15;

    const QParams& p = P.q[q];
    const float* xc   = ws + (size_t)q * PERQ + 2 * QF;
    const float* xdbl = ws + (size_t)q * PERQ + 3 * QF;
    float*       y    = ws + (size_t)q * PERQ;         // alias over x_in
    const size_t rb   = (size_t)b * LQ;

    const float A   = -__expf(p.A_log[e * D_STATE + n]);
    const float dtb = p.dt_bias[e];
    const float Dp  = p.D[e];
    float dtW[DT_RANK];
    #pragma unroll
    for (int j = 0; j < DT_RANK; ++j) dtW[j] = p.dt_proj[j * D_INNER + e];

#if defined(HAVE_TDM)
    // issue both tile DMAs for chunk `c` into buffer `buf`
    auto issue = [&](int l0, int buf) {
        unsigned long long gxd = (unsigned long long)(xdbl + (rb + l0) * XDP);
        unsigned long long gxc = (unsigned long long)(xc + (rb + l0) * D_INNER + e0);
        unsigned lds_xd = (unsigned)(unsigned long long)(&xd_s[buf][0]);
        unsigned lds_xc = (unsigned)(unsigned long long)(&xc_s[buf][0]);
        // x_dbl chunk: contiguous 32*48 floats (1-D descriptor)
        tdm_load(tdm_g0(lds_xd, gxd), tdm_g1(SCH * XDP, 1, SCH * XDP, 0, SCH * XDP));
        // xc chunk: 32 rows x 8 floats, row stride 192 (2-D descriptor)
        tdm_load(tdm_g0(lds_xc, gxc), tdm_g1(8, SCH, 8, SCH, D_INNER));
    };
    if (t < 32) {
        issue(0, 0);
        __builtin_amdgcn_s_wait_tensorcnt(0);
    }
    asm volatile("" ::: "memory");
    __syncthreads();
#else
    {
        const float* src = xdbl + rb * XDP;
        for (int idx = t; idx < SCH * XDP; idx += 128) xd_s[0][idx] = src[idx];
        for (int idx = t; idx < SCH * 8; idx += 128) {
            const int r = idx >> 3, c = idx & 7;
            xc_s[0][idx] = xc[(rb + r) * D_INNER + e0 + c];
        }
        __syncthreads();
    }
#endif

    float h = 0.f;
    const int nchunks = LQ / SCH;
    for (int ci = 0; ci < nchunks; ++ci) {
        const int l0  = ci * SCH;
        const int cur = ci & 1;

#if defined(HAVE_TDM)
        if (ci + 1 < nchunks && t < 32) issue(l0 + SCH, cur ^ 1);
#else
        if (ci + 1 < nchunks) {
            const float* src = xdbl + (rb + l0 + SCH) * XDP;
            for (int idx = t; idx < SCH * XDP; idx += 128) xd_s[cur ^ 1][idx] = src[idx];
            for (int idx = t; idx < SCH * 8; idx += 128) {
                const int r = idx >> 3, c = idx & 7;
                xc_s[cur ^ 1][idx] = xc[(rb + l0 + SCH + r) * D_INNER + e0 + c];
            }
        }
#endif

        for (int s = 0; s < SCH; ++s) {
            const float* row = &xd_s[cur][s * XDP];
            float dta = dtb;
            #pragma unroll
            for (int j = 0; j < DT_RANK; ++j) dta = fmaf(row[j], dtW[j], dta);
            const float dt = (dta > 0.f) ? dta + __logf(1.f + __expf(-dta))
                                         : __logf(1.f + __expf(dta));
            const float xt = xc_s[cur][s * 8 + ch];
            const float Bv = row[DT_RANK + n];
            const float Cv = row[DT_RANK + D_STATE + n];
            h = fmaf(__expf(dt * A), h, dt * xt * Bv);
            float yv = h * Cv;
            yv += __shfl_xor(yv, 8, 16);
            yv += __shfl_xor(yv, 4, 16);
            yv += __shfl_xor(yv, 2, 16);
            yv += __shfl_xor(yv, 1, 16);
            if (n == 0) y_s[s * 8 + ch] = fmaf(Dp, xt, yv);
        }
        __syncthreads();                    // y_s complete

        for (int idx = t; idx < SCH * 8; idx += 128) {
            const int r = idx >> 3, c = idx & 7;
            y[(rb + l0 + r) * D_INNER + e0 + c] = y_s[idx];
        }

#if defined(HAVE_TDM)
        if (t < 32) __builtin_amdgcn_s_wait_tensorcnt(0);
        asm volatile("" ::: "memory");
#endif
        __syncthreads();                    // next buffer ready; y_s reusable
    }
}

// ---------------------------------------------------------------------------
// K4: out = (y * silu(res)) @ out_proj, scattered into quadrant of NCHW out.
// grid = (8192/16, 4), block 32.  One wave computes the full 16x96 row block:
// the silu-gated A-fragment is computed once and reused by 6 WMMA tiles.
// ---------------------------------------------------------------------------
__global__ __launch_bounds__(32)
void k_out_proj(AllParams P, const float* __restrict__ ws, float* __restrict__ out) {
    const int q    = blockIdx.y;
    const int lane = threadIdx.x;
    const int m0   = blockIdx.x * 16;

    const QParams& p = P.q[q];
    const float* y   = ws + (size_t)q * PERQ;        // aliased x_in
    const float* res = y + QF;
    const float* W   = p.out_proj;                   // (192,96)

    const int colb = lane & 15;
    const int rowA = m0 + (lane & 15);
    const int kk   = (lane >> 4) * 2;

    const float* ya = y   + (size_t)rowA * D_INNER;
    const float* ra = res + (size_t)rowA * D_INNER;

    v8f acc[6];
    #pragma unroll
    for (int j = 0; j < 6; ++j) acc[j] = (v8f){0.f,0.f,0.f,0.f,0.f,0.f,0.f,0.f};

    #pragma unroll 2
    for (int k = 0; k < D_INNER; k += 4) {
        v2f a;
        const float y0 = ya[k + kk],     r0 = ra[k + kk];
        const float y1 = ya[k + kk + 1], r1 = ra[k + kk + 1];
        a.x = y0 * silu_f(r0);
        a.y = y1 * silu_f(r1);
        const float* w0 = W + (k + kk) * D_MODEL + colb;
        #pragma unroll
        for (int j = 0; j < 6; ++j) {
            v2f b;
            b.x = w0[16 * j];
            b.y = w0[D_MODEL + 16 * j];
            acc[j] = __builtin_amdgcn_wmma_f32_16x16x4_f32(
                         false, a, false, b, (short)0, acc[j], false, false);
        }
    }

    const int oh = (q >> 1) * QH, ow = (q & 1) * QW;
    const int mBase = m0 + ((lane >> 4) << 3);
    #pragma unroll
    for (int j = 0; j < 6; ++j) {
        const int col = colb + 16 * j;
        #pragma unroll
        for (int r = 0; r < 8; ++r) {
            const int m  = mBase + r;
            const int bb = m >> 12;
            const int l  = m & (LQ - 1);
            out[(((size_t)bb * D_MODEL + col) * 128 + oh + (l >> 6)) * 128
                + ow + (l & 63)] = acc[j][r];
        }
    }
}

// ---------------------------------------------------------------------------
// launch
// ---------------------------------------------------------------------------
extern "C" void kernel_launch(void* const* d_in, const int* in_sizes, int n_in,
                              void* d_out, int out_size, void* d_ws, size_t ws_size,
                              hipStream_t stream) {
    const float* x = (const float*)d_in[0];
    AllParams P;
    for (int q = 0; q < 4; ++q) {
        const int base = 1 + q * 9;
        P.q[q].in_proj  = (const float*)d_in[base + 0];
        P.q[q].conv_w   = (const float*)d_in[base + 1];
        P.q[q].conv_b   = (const float*)d_in[base + 2];
        P.q[q].x_proj   = (const float*)d_in[base + 3];
        P.q[q].dt_proj  = (const float*)d_in[base + 4];
        P.q[q].dt_bias  = (const float*)d_in[base + 5];
        P.q[q].A_log    = (const float*)d_in[base + 6];
        P.q[q].D        = (const float*)d_in[base + 7];
        P.q[q].out_proj = (const float*)d_in[base + 8];
    }
    float* ws  = (float*)d_ws;
    float* out = (float*)d_out;

    dim3 g1(2 * D_INNER / 64, MROWS / 16, 4);
    k_in_proj<<<g1, 32, 0, stream>>>(x, P, ws);

    dim3 g2(LQ / CT, BATCH, 4);
    k_conv_xproj<<<g2, 256, 0, stream>>>(P, ws);

    dim3 g3(D_INNER / 8, BATCH, 4);
    k_scan<<<g3, 128, 0, stream>>>(P, ws);

    dim3 g4(MROWS / 16, 4);
    k_out_proj<<<g4, 32, 0, stream>>>(P, ws, out);
}